// GGNN_5325759447713
// MI455X (gfx1250) — compile-verified
//
#include <hip/hip_runtime.h>
#include <hip/hip_bf16.h>
#include <math.h>

typedef float v2f __attribute__((ext_vector_type(2)));
typedef float v8f __attribute__((ext_vector_type(8)));

#define KDIM 128

// ---------------------------------------------------------------------------
// Tiled GEMM using V_WMMA_F32_16X16X4_F32.
//   Y[:, cbase:cbase+ccols] = act(X) @ B (+ bias),  B is KDIM x ccols
// TRANS_FILL = false : W is [out_cols_total][KDIM] row-major (X @ W.T case)
// TRANS_FILL = true  : W is [KDIM][out_cols_total] row-major (X @ W case)
// RELU_IN applies relu to the A operand as it is loaded.
// 256 threads = 8 waves; each wave computes a 16-row x 128-col stripe.
// ---------------------------------------------------------------------------
template <bool TRANS_FILL, bool RELU_IN>
__global__ __launch_bounds__(256) void ggnn_gemm(
    const float* __restrict__ X, const float* __restrict__ W,
    const float* __restrict__ bias, float* __restrict__ Y,
    int nrows, int out_stride, int out_cols_total)
{
    __shared__ float Wt[KDIM * KDIM];   // Wt[n][k] = B[k][n], 64 KB

    const int tid   = threadIdx.x;
    const int cbase = blockIdx.y * KDIM;
    int ccols = out_cols_total - cbase;
    if (ccols > KDIM) ccols = KDIM;

    // Stage weight chunk into LDS.
    for (int i = tid; i < KDIM * KDIM; i += 256) {
        const int n = i >> 7;
        const int k = i & (KDIM - 1);
        float w = 0.0f;
        if (n < ccols) {
            if (TRANS_FILL) w = W[(size_t)k * out_cols_total + cbase + n];
            else            w = W[(size_t)(cbase + n) * KDIM + k];
        }
        Wt[n * KDIM + k] = w;
    }
    __syncthreads();

    const int wave = tid >> 5;
    const int lane = tid & 31;
    const int m0 = (blockIdx.x * 8 + wave) * 16;
    if (m0 >= nrows) return;

    const int half = lane >> 4;   // 0: K pair (k,k+1); 1: K pair (k+2,k+3)
    const int l16  = lane & 15;

    const float* Xrow = X + (size_t)(m0 + l16) * KDIM + half * 2;

    v8f acc[8] = {};

    for (int kb = 0; kb < KDIM; kb += 4) {
        v2f a = *(const v2f*)(Xrow + kb);
        if (RELU_IN) { a.x = fmaxf(a.x, 0.0f); a.y = fmaxf(a.y, 0.0f); }
        #pragma unroll
        for (int t = 0; t < 8; ++t) {
            v2f b = *(const v2f*)(&Wt[(t * 16 + l16) * KDIM + kb + half * 2]);
            acc[t] = __builtin_amdgcn_wmma_f32_16x16x4_f32(
                false, a, false, b, (short)0, acc[t], false, false);
        }
    }

    // D layout: VGPR r -> row m0 + r (+8 for lanes 16..31), col = l16.
    #pragma unroll
    for (int t = 0; t < 8; ++t) {
        const int c = t * 16 + l16;
        if (c < ccols) {
            const float bv = bias ? bias[cbase + c] : 0.0f;
            #pragma unroll
            for (int r = 0; r < 8; ++r) {
                const int row = m0 + r + half * 8;
                Y[(size_t)row * out_stride + cbase + c] = acc[t][r] + bv;
            }
        }
    }
}

// ---------------------------------------------------------------------------
// agg[dst[e]] += m[src[e]] ; one wave per edge, float4 per lane (32*4 = 128).
// ---------------------------------------------------------------------------
__global__ __launch_bounds__(256) void ggnn_scatter(
    const float* __restrict__ m, const int* __restrict__ src,
    const int* __restrict__ dst, float* __restrict__ agg, int nedges)
{
    const long idx = (long)blockIdx.x * 256 + threadIdx.x;
    const long e = idx >> 5;
    if (e >= nedges) return;
    const int lane = (int)(idx & 31);
    const int s = src[e];
    const int d = dst[e];
    const float4 v = *(const float4*)(m + (size_t)s * KDIM + lane * 4);
    float* out = agg + (size_t)d * KDIM + lane * 4;
    atomicAdd(out + 0, v.x);
    atomicAdd(out + 1, v.y);
    atomicAdd(out + 2, v.z);
    atomicAdd(out + 3, v.w);
}

__global__ __launch_bounds__(256) void ggnn_zero(float4* __restrict__ p, long n4)
{
    const long i = (long)blockIdx.x * 256 + threadIdx.x;
    if (i < n4) p[i] = make_float4(0.f, 0.f, 0.f, 0.f);
}

// ---------------------------------------------------------------------------
// GRU gate fusion: h = (1-z)*n + z*h   (in-place on h; per-node independent)
// ---------------------------------------------------------------------------
__global__ __launch_bounds__(256) void ggnn_gru_gates(
    const float* __restrict__ gi, const float* __restrict__ gh,
    float* __restrict__ h, int nnodes)
{
    const long idx = (long)blockIdx.x * 256 + threadIdx.x;
    if (idx >= (long)nnodes * KDIM) return;
    const long n = idx >> 7;
    const int  c = (int)(idx & (KDIM - 1));
    const float* gin = gi + n * 384;
    const float* ghn = gh + n * 384;
    const float i_r = gin[c], i_z = gin[c + 128], i_n = gin[c + 256];
    const float h_r = ghn[c], h_z = ghn[c + 128], h_n = ghn[c + 256];
    const float r = 1.0f / (1.0f + __expf(-(i_r + h_r)));
    const float z = 1.0f / (1.0f + __expf(-(i_z + h_z)));
    const float nn = tanhf(i_n + r * h_n);
    const float hv = h[idx];
    h[idx] = (1.0f - z) * nn + z * hv;
}

// ---------------------------------------------------------------------------
extern "C" void kernel_launch(void* const* d_in, const int* in_sizes, int n_in,
                              void* d_out, int out_size, void* d_ws, size_t ws_size,
                              hipStream_t stream)
{
    (void)n_in; (void)out_size; (void)ws_size;

    const float* x      = (const float*)d_in[0];
    const int*   ei     = (const int*)d_in[1];
    const float* lin_w  = (const float*)d_in[2];
    const float* lin_b  = (const float*)d_in[3];
    const float* ggc_w  = (const float*)d_in[4];
    const float* w_ih   = (const float*)d_in[5];
    const float* w_hh   = (const float*)d_in[6];
    const float* b_ih   = (const float*)d_in[7];
    const float* b_hh   = (const float*)d_in[8];
    const float* lin2_w = (const float*)d_in[9];
    const float* lin2_b = (const float*)d_in[10];
    float* out = (float*)d_out;

    const int N = in_sizes[0] / KDIM;
    const int E = in_sizes[1] / 2;
    const int* src = ei;       // edge_index[0]
    const int* dst = ei + E;   // edge_index[1]

    float* h   = (float*)d_ws;
    float* m   = h   + (size_t)N * 128;
    float* agg = m   + (size_t)N * 128;
    float* gi  = agg + (size_t)N * 128;
    float* gh  = gi  + (size_t)N * 384;

    const int row_tiles  = (N + 15) / 16;
    const int row_blocks = (row_tiles + 7) / 8;
    const dim3 blk(256);

    const long agg4 = (long)N * 32;                       // float4 count
    const unsigned zero_blks    = (unsigned)((agg4 + 255) / 256);
    const unsigned scatter_blks = (unsigned)(((long)E * 32 + 255) / 256);
    const unsigned gate_blks    = (unsigned)(((long)N * 128 + 255) / 256);

    // h = x @ lin_w.T + lin_b
    ggnn_gemm<false, false><<<dim3(row_blocks, 1), blk, 0, stream>>>(
        x, lin_w, lin_b, h, N, 128, 128);

    for (int l = 0; l < 8; ++l) {
        // m = h @ ggc_w[l]
        ggnn_gemm<true, false><<<dim3(row_blocks, 1), blk, 0, stream>>>(
            h, ggc_w + (size_t)l * 128 * 128, nullptr, m, N, 128, 128);
        // agg = segment_sum(m[src], dst)
        ggnn_zero<<<zero_blks, blk, 0, stream>>>((float4*)agg, agg4);
        ggnn_scatter<<<scatter_blks, blk, 0, stream>>>(m, src, dst, agg, E);
        // gi = agg @ w_ih.T + b_ih ; gh = h @ w_hh.T + b_hh   (384 cols -> grid.y = 3)
        ggnn_gemm<false, false><<<dim3(row_blocks, 3), blk, 0, stream>>>(
            agg, w_ih, b_ih, gi, N, 384, 384);
        ggnn_gemm<false, false><<<dim3(row_blocks, 3), blk, 0, stream>>>(
            h, w_hh, b_hh, gh, N, 384, 384);
        // gates, in-place h update
        ggnn_gru_gates<<<gate_blks, blk, 0, stream>>>(gi, gh, h, N);
    }

    // out = relu(h) @ lin2_w.T + lin2_b  (100 cols, guarded)
    ggnn_gemm<false, true><<<dim3(row_blocks, 1), blk, 0, stream>>>(
        h, lin2_w, lin2_b, out, N, 100, 100);
}